// DbrxRouter_14955076125244
// MI455X (gfx1250) — compile-verified
//
#include <hip/hip_runtime.h>
#include <hip/hip_bf16.h>
#include <math.h>

typedef float v2f __attribute__((ext_vector_type(2)));
typedef float v8f __attribute__((ext_vector_type(8)));

#define D_DIM 6144
#define E_DIM 16
#define TOPK 4
#define SPLITK_WAVES 8                     // 8 waves per block, each owns D/8 = 768 of K
#define K_SLICE (D_DIM / SPLITK_WAVES)
#define K_ITERS (K_SLICE / 4)              // 192 WMMAs per wave, compile-time constant

// One block (8 waves) handles a 16-token x 16-expert tile with split-K.
// Each wave accumulates its K-slice via V_WMMA_F32_16X16X4_F32, partials are
// reduced in LDS, wave 0 runs softmax + top-4 + L1 renorm.
__global__ __launch_bounds__(32 * SPLITK_WAVES)
void dbrx_router_kernel(const float* __restrict__ x,
                        const float* __restrict__ W,
                        float* __restrict__ out_w,
                        float* __restrict__ out_tw,
                        int* __restrict__ out_te)
{
    __shared__ float cpart[SPLITK_WAVES * 32 * 8];   // 8 KB

    const int lane = threadIdx.x & 31;
    const int wave = threadIdx.x >> 5;
    const int tile = blockIdx.x;                     // one tile per block

    const int half = lane >> 4;                      // 0: lanes 0-15, 1: lanes 16-31
    const int lq   = lane & 15;

    const int k0 = wave * K_SLICE;                   // folded into pointers below

    // A (16x4 f32): lane half h, element j holds K = 2h + j -> contiguous float2/lane
    const float* xrow = x + (size_t)(tile * 16 + lq) * D_DIM + k0 + (half * 2);
    // B (4x16 f32): VGPR j, lane half h holds K = 2h + j, N = lq (row-major W[d][e])
    const float* wcol = W + ((size_t)k0 + half * 2) * E_DIM + lq;

    v8f c = {};
    // Constant trip count -> pure SALU loop control, EXEC stays all-1s for WMMA.
    #pragma unroll 8
    for (int i = 0; i < K_ITERS; ++i) {
        // x is streamed once: non-temporal so it doesn't evict W from L2/WGP$
        v2f a = __builtin_nontemporal_load((const v2f*)(xrow + i * 4));
        v2f b;
        b.x = wcol[(size_t)(i * 4) * E_DIM];         // W stays cache-resident (RT)
        b.y = wcol[(size_t)(i * 4) * E_DIM + E_DIM];
        // (neg_a, A, neg_b, B, c_mod, C, reuse_a, reuse_b)
        c = __builtin_amdgcn_wmma_f32_16x16x4_f32(false, a, false, b,
                                                  (short)0, c, false, false);
    }

    // ---- split-K reduction through LDS ----
    float* myp = &cpart[(wave * 32 + lane) * 8];
    #pragma unroll
    for (int v = 0; v < 8; ++v) myp[v] = c[v];
    __syncthreads();
    if (wave != 0) return;                           // wave-uniform exit

    float acc[8];
    #pragma unroll
    for (int v = 0; v < 8; ++v) acc[v] = 0.0f;
    #pragma unroll
    for (int w = 0; w < SPLITK_WAVES; ++w) {
        const float* p = &cpart[(w * 32 + lane) * 8];
        #pragma unroll
        for (int v = 0; v < 8; ++v) acc[v] += p[v];
    }

    // C/D layout: VGPR v, lane half h -> row (v + 8h), expert lq.
    const size_t base_row = (size_t)tile * 16;

    // ---- softmax over 16 experts (16-lane group reductions; xor masks <= 8
    //      never cross the half-wave boundary) ----
    float wn[8];
    #pragma unroll
    for (int v = 0; v < 8; ++v) {
        float lv = acc[v];
        float m = lv;
        #pragma unroll
        for (int off = 8; off >= 1; off >>= 1)
            m = fmaxf(m, __shfl_xor(m, off, 32));
        float e = __expf(lv - m);
        float s = e;
        #pragma unroll
        for (int off = 8; off >= 1; off >>= 1)
            s += __shfl_xor(s, off, 32);
        wn[v] = e / s;
    }

    // ---- store full softmax weights [token][16] (write-once -> NT store) ----
    #pragma unroll
    for (int v = 0; v < 8; ++v) {
        size_t row = base_row + v + half * 8;
        __builtin_nontemporal_store(wn[v], &out_w[row * E_DIM + lq]);
    }

    // ---- top-4 with L1 renorm ----
    #pragma unroll
    for (int v = 0; v < 8; ++v) {
        size_t row = base_row + v + half * 8;
        float val = wn[v];
        float tv[TOPK];
        int   ti[TOPK];
        #pragma unroll
        for (int j = 0; j < TOPK; ++j) {
            float bv = val;
            int   bi = lq;
            #pragma unroll
            for (int off = 8; off >= 1; off >>= 1) {
                float ov = __shfl_xor(bv, off, 32);
                int   oi = __shfl_xor(bi, off, 32);
                if (ov > bv || (ov == bv && oi < bi)) { bv = ov; bi = oi; }
            }
            tv[j] = bv;
            ti[j] = bi;
            if (lq == bi) val = -INFINITY;           // remove winner for next round
        }
        float norm = tv[0] + tv[1] + tv[2] + tv[3];
        if (lq < TOPK) {
            float sv = tv[lq];
            int   si = ti[lq];
            __builtin_nontemporal_store(sv / norm, &out_tw[row * TOPK + lq]);
            __builtin_nontemporal_store(si, &out_te[row * TOPK + lq]);
        }
    }
}

extern "C" void kernel_launch(void* const* d_in, const int* in_sizes, int n_in,
                              void* d_out, int out_size, void* d_ws, size_t ws_size,
                              hipStream_t stream) {
    const float* x = (const float*)d_in[0];
    const float* W = (const float*)d_in[1];

    const int n_tokens = in_sizes[0] / D_DIM;        // 4*4096 = 16384
    const int n_tiles  = (n_tokens + 15) / 16;       // 1024 -> one block per tile

    float* out    = (float*)d_out;
    float* out_w  = out;                                            // [tokens,16]
    float* out_tw = out + (size_t)n_tokens * E_DIM;                 // [tokens,4]
    int*   out_te = (int*)(out + (size_t)n_tokens * E_DIM
                               + (size_t)n_tokens * TOPK);          // [tokens,4] i32

    dbrx_router_kernel<<<n_tiles, 32 * SPLITK_WAVES, 0, stream>>>(
        x, W, out_w, out_tw, out_te);
}